// NEGCN_25615184953980
// MI455X (gfx1250) — compile-verified
//
#include <hip/hip_runtime.h>
#include <math.h>

// Problem constants (match reference)
#define BB 64
#define TT 512
#define DD 256
#define EPSV 1e-6f
#define LNEPS 1e-5f

typedef __attribute__((ext_vector_type(2))) float v2f;
typedef __attribute__((ext_vector_type(8))) float v8f;

// D = A(16x4, f32) * B(4x16, f32) + C(16x16, f32)   -- exact fp32 matrix op
__device__ __forceinline__ v8f wmma4(v2f a, v2f b, v8f c) {
    return __builtin_amdgcn_wmma_f32_16x16x4_f32(
        /*neg_a=*/false, a, /*neg_b=*/false, b,
        /*c_mod=*/(short)0, c, /*reuse_a=*/false, /*reuse_b=*/false);
}

// ---- block-wide sum over 256 threads (wave32: 8 waves) ----
__device__ __forceinline__ float blockSum256(float v) {
    __shared__ float red[8];
    #pragma unroll
    for (int o = 16; o > 0; o >>= 1) v += __shfl_down(v, o, 32);
    const int lane = threadIdx.x & 31, w = threadIdx.x >> 5;
    if (lane == 0) red[w] = v;
    __syncthreads();
    if (threadIdx.x == 0) {
        float t = red[0];
        #pragma unroll
        for (int i = 1; i < 8; ++i) t += red[i];
        red[0] = t;
    }
    __syncthreads();
    float t = red[0];
    __syncthreads();   // allow reuse of red by a second call
    return t;
}

// ---------------- 1) xn = x / max(||x||, 1e-12) : one block per row ----------------
__global__ __launch_bounds__(256) void k_rownorm(const float* __restrict__ x,
                                                 float* __restrict__ xn) {
    const size_t row = blockIdx.x;          // B*T rows
    const float v = x[row * DD + threadIdx.x];
    const float ss = blockSum256(v * v);
    const float nrm = fmaxf(sqrtf(ss), 1e-12f);
    xn[row * DD + threadIdx.x] = v / nrm;
}

// ---------------- 2) corel: 32x32 region per wave, 2x2 WMMA accumulators ----------
// corel = 0.5*mm*(xn.xn^T + sl_sim), band -> -1
__global__ __launch_bounds__(256) void k_corel(const float* __restrict__ xn,
                                               const float* __restrict__ sl,
                                               const float* __restrict__ mask,
                                               float* __restrict__ corel) {
    const int lane = threadIdx.x & 31;
    const int reg  = blockIdx.x * 8 + (threadIdx.x >> 5);
    const int regRow   = TT / 32;                        // 16
    const int regBatch = regRow * regRow;                // 256
    const int b  = reg / regBatch;
    const int r  = reg % regBatch;
    const int tr = (r / regRow) * 32;                    // t (row) region base
    const int tc = (r % regRow) * 32;                    // s (col) region base

    const float* Xb = xn + (size_t)b * TT * DD;
    const int l15 = lane & 15;
    const int kq  = (lane >> 4) * 2;

    const float* ar0 = Xb + (size_t)(tr + l15) * DD;
    const float* ar1 = ar0 + (size_t)16 * DD;
    const float* br0 = Xb + (size_t)(tc + l15) * DD;     // B = A^T -> same pattern
    const float* br1 = br0 + (size_t)16 * DD;

    v8f acc[2][2] = {{{}, {}}, {{}, {}}};
    #pragma unroll 4
    for (int k = 0; k < DD; k += 4) {
        const v2f A0 = *(const v2f*)(ar0 + k + kq);
        const v2f A1 = *(const v2f*)(ar1 + k + kq);
        const v2f B0 = *(const v2f*)(br0 + k + kq);
        const v2f B1 = *(const v2f*)(br1 + k + kq);
        acc[0][0] = wmma4(A0, B0, acc[0][0]);
        acc[0][1] = wmma4(A0, B1, acc[0][1]);
        acc[1][0] = wmma4(A1, B0, acc[1][0]);
        acc[1][1] = wmma4(A1, B1, acc[1][1]);
    }

    const float* slb = sl + (size_t)b * TT;
    const float* mb  = mask + (size_t)b * TT;
    #pragma unroll
    for (int j = 0; j < 2; ++j) {
        const int ncol = tc + j * 16 + l15;
        const float sc = fminf(fmaxf(slb[ncol], 0.f), 1.f);
        const float mc = mb[ncol];
        #pragma unroll
        for (int i = 0; i < 2; ++i) {
            const int rbase = tr + i * 16 + ((lane >> 4) << 3);
            #pragma unroll
            for (int v = 0; v < 8; ++v) {
                const int row = rbase + v;
                const float sr = fminf(fmaxf(slb[row], 0.f), 1.f);
                const float mm = mb[row] * mc;
                const float slsim = (1.0f - fabsf(sc - sr)) * 0.5f * (sc * sc + sr * sr);
                float val = 0.5f * (acc[i][j][v] + slsim) * mm;
                int dd = row - ncol; if (dd < 0) dd = -dd;
                if (dd <= 1) val = -1.0f;                // band override
                corel[((size_t)b * TT + row) * TT + ncol] = val;
            }
        }
    }
}

// ---------------- 3) top-2 indices per row (first-occurrence ties, like lax.top_k) --
__global__ __launch_bounds__(256) void k_topk(const float* __restrict__ corel,
                                              int* __restrict__ inds) {
    const size_t row = (size_t)blockIdx.x * blockDim.x + threadIdx.x;
    if (row >= (size_t)BB * TT) return;
    const float* c = corel + row * TT;
    float v1 = -INFINITY, v2 = -INFINITY;
    int   i1 = 0, i2 = 0;
    for (int s = 0; s < TT; ++s) {
        const float v = c[s];
        if (v > v1)      { v2 = v1; i2 = i1; v1 = v; i1 = s; }
        else if (v > v2) { v2 = v;  i2 = s; }
    }
    inds[row * 2 + 0] = i1;
    inds[row * 2 + 1] = i2;
}

// ---------------- 4) adj = clip(base+ext,0,1)*corel + I   (in-place over corel) ----
__global__ __launch_bounds__(256) void k_adjbuild(const float* __restrict__ mask,
                                                  const int* __restrict__ inds,
                                                  float* __restrict__ adj) {
    const size_t idx = (size_t)blockIdx.x * blockDim.x + threadIdx.x;  // B*T*T
    const int s = (int)(idx % TT);
    const size_t r = idx / TT;
    const int t = (int)(r % TT);
    const int b = (int)(r / TT);
    const float c = adj[idx];
    const int* it = inds + ((size_t)b * TT + t) * 2;
    const int* is = inds + ((size_t)b * TT + s) * 2;
    const float mt = mask[(size_t)b * TT + t], ms = mask[(size_t)b * TT + s];
    const float e1 = (it[0] == s || it[1] == s) ? 1.f : 0.f;
    const float e2 = (is[0] == t || is[1] == t) ? 1.f : 0.f;
    const float ext = fmaxf(e1, e2) * mt * ms;
    int dd = t - s; if (dd < 0) dd = -dd;
    const float base = (dd <= 1) ? 1.f : 0.f;
    float a = fminf(fmaxf(base + ext, 0.f), 1.f) * c;
    if (t == s) a += 1.f;
    adj[idx] = a;
}

// ---------------- 5) row & column sums (+EPS) ----------------
__global__ __launch_bounds__(256) void k_rowsum(const float* __restrict__ adj,
                                                float* __restrict__ rows) {
    const size_t row = blockIdx.x;                       // B*T
    const float* a = adj + row * TT;
    const float s = blockSum256(a[threadIdx.x] + a[threadIdx.x + 256]);
    if (threadIdx.x == 0) rows[row] = s + EPSV;
}

__global__ __launch_bounds__(256) void k_colsum(const float* __restrict__ adj,
                                                float* __restrict__ cols) {
    const int b = blockIdx.x / TT, c = blockIdx.x % TT;
    const float* a = adj + (size_t)b * TT * TT + c;
    float s = 0.f;
    for (int t = threadIdx.x; t < TT; t += 256) s += a[(size_t)t * TT];
    s = blockSum256(s);
    if (threadIdx.x == 0) cols[blockIdx.x] = s + EPSV;
}

// ---------------- 6) adj = adj / sqrt(row*col) * mask_mat ----------------
__global__ __launch_bounds__(256) void k_adjnorm(float* __restrict__ adj,
                                                 const float* __restrict__ rows,
                                                 const float* __restrict__ cols,
                                                 const float* __restrict__ mask) {
    const size_t idx = (size_t)blockIdx.x * blockDim.x + threadIdx.x;
    const int s = (int)(idx % TT);
    const size_t r = idx / TT;
    const int t = (int)(r % TT);
    const int b = (int)(r / TT);
    float v = adj[idx];
    v *= rsqrtf(rows[(size_t)b * TT + t] * cols[(size_t)b * TT + s]);
    v *= mask[(size_t)b * TT + t] * mask[(size_t)b * TT + s];
    adj[idx] = v;
}

// ---------------- 7) H = adj @ X : 32x32 region per wave, 2x2 accumulators --------
__global__ __launch_bounds__(256) void k_adjx(const float* __restrict__ adj,
                                              const float* __restrict__ X,
                                              float* __restrict__ H) {
    const int lane = threadIdx.x & 31;
    const int reg  = blockIdx.x * 8 + (threadIdx.x >> 5);
    const int regN     = DD / 32;                        // 8
    const int regBatch = (TT / 32) * regN;               // 128
    const int b  = reg / regBatch;
    const int r  = reg % regBatch;
    const int tr = (r / regN) * 32;
    const int tc = (r % regN) * 32;

    const float* A  = adj + (size_t)b * TT * TT;
    const float* Xb = X   + (size_t)b * TT * DD;
    const int l15 = lane & 15;
    const int kq  = (lane >> 4) * 2;

    const float* ar0 = A + (size_t)(tr + l15) * TT;
    const float* ar1 = ar0 + (size_t)16 * TT;
    const int ncol0 = tc + l15;
    const int ncol1 = ncol0 + 16;

    v8f acc[2][2] = {{{}, {}}, {{}, {}}};
    #pragma unroll 4
    for (int k = 0; k < TT; k += 4) {
        const v2f A0 = *(const v2f*)(ar0 + k + kq);
        const v2f A1 = *(const v2f*)(ar1 + k + kq);
        const float* xr0 = Xb + (size_t)(k + kq) * DD;
        const float* xr1 = xr0 + DD;
        v2f B0; B0.x = xr0[ncol0]; B0.y = xr1[ncol0];
        v2f B1; B1.x = xr0[ncol1]; B1.y = xr1[ncol1];
        acc[0][0] = wmma4(A0, B0, acc[0][0]);
        acc[0][1] = wmma4(A0, B1, acc[0][1]);
        acc[1][0] = wmma4(A1, B0, acc[1][0]);
        acc[1][1] = wmma4(A1, B1, acc[1][1]);
    }
    #pragma unroll
    for (int i = 0; i < 2; ++i) {
        const int rbase = tr + i * 16 + ((lane >> 4) << 3);
        #pragma unroll
        for (int v = 0; v < 8; ++v) {
            H[((size_t)b * TT + rbase + v) * DD + ncol0] = acc[i][0][v];
            H[((size_t)b * TT + rbase + v) * DD + ncol1] = acc[i][1][v];
        }
    }
}

// ---------------- 8) Hout = ELU(Hin @ W^T + bias) : 32x32 region per wave ---------
__global__ __launch_bounds__(256) void k_gemmwt(const float* __restrict__ Hin,
                                                const float* __restrict__ W,
                                                const float* __restrict__ bias,
                                                float* __restrict__ Hout) {
    const int lane = threadIdx.x & 31;
    const int reg  = blockIdx.x * 8 + (threadIdx.x >> 5);
    const int regN = DD / 32;                            // 8
    const int tr = (reg / regN) * 32;                    // row in M = B*T
    const int tc = (reg % regN) * 32;

    const int l15 = lane & 15;
    const int kq  = (lane >> 4) * 2;

    const float* ar0 = Hin + (size_t)(tr + l15) * DD;
    const float* ar1 = ar0 + (size_t)16 * DD;
    const float* br0 = W + (size_t)(tc + l15) * DD;      // B[k][n] = W[n][k]
    const float* br1 = br0 + (size_t)16 * DD;

    v8f acc[2][2] = {{{}, {}}, {{}, {}}};
    #pragma unroll 4
    for (int k = 0; k < DD; k += 4) {
        const v2f A0 = *(const v2f*)(ar0 + k + kq);
        const v2f A1 = *(const v2f*)(ar1 + k + kq);
        const v2f B0 = *(const v2f*)(br0 + k + kq);
        const v2f B1 = *(const v2f*)(br1 + k + kq);
        acc[0][0] = wmma4(A0, B0, acc[0][0]);
        acc[0][1] = wmma4(A0, B1, acc[0][1]);
        acc[1][0] = wmma4(A1, B0, acc[1][0]);
        acc[1][1] = wmma4(A1, B1, acc[1][1]);
    }
    #pragma unroll
    for (int j = 0; j < 2; ++j) {
        const int ncol = tc + j * 16 + l15;
        const float bn = bias[ncol];
        #pragma unroll
        for (int i = 0; i < 2; ++i) {
            const int rbase = tr + i * 16 + ((lane >> 4) << 3);
            #pragma unroll
            for (int v = 0; v < 8; ++v) {
                float h = acc[i][j][v] + bn;
                h = (h > 0.f) ? h : expm1f(h);           // ELU(alpha=1)
                Hout[(size_t)(rbase + v) * DD + ncol] = h;
            }
        }
    }
}

// ---------------- 9) LayerNorm over D, * g + be ----------------
__global__ __launch_bounds__(256) void k_layernorm(const float* __restrict__ Hin,
                                                   const float* __restrict__ g,
                                                   const float* __restrict__ be,
                                                   float* __restrict__ out) {
    const size_t row = blockIdx.x;                       // B*T
    const float v = Hin[row * DD + threadIdx.x];
    const float mu = blockSum256(v) * (1.0f / DD);
    const float d  = v - mu;
    const float var = blockSum256(d * d) * (1.0f / DD);
    out[row * DD + threadIdx.x] =
        d * rsqrtf(var + LNEPS) * g[threadIdx.x] + be[threadIdx.x];
}

extern "C" void kernel_launch(void* const* d_in, const int* in_sizes, int n_in,
                              void* d_out, int out_size, void* d_ws, size_t ws_size,
                              hipStream_t stream) {
    (void)in_sizes; (void)n_in; (void)out_size; (void)ws_size;
    const float* x    = (const float*)d_in[0];
    const float* sl   = (const float*)d_in[1];
    const float* mask = (const float*)d_in[2];
    const float* W1   = (const float*)d_in[3];
    const float* b1   = (const float*)d_in[4];
    const float* g1   = (const float*)d_in[5];
    const float* be1  = (const float*)d_in[6];
    const float* W2   = (const float*)d_in[7];
    const float* b2   = (const float*)d_in[8];
    const float* g2   = (const float*)d_in[9];
    const float* be2  = (const float*)d_in[10];
    float* out = (float*)d_out;

    // workspace layout (floats): xn | adj | inds | rows | cols | hA | hB | h1
    float* ws   = (float*)d_ws;
    float* xn   = ws;                                   // B*T*D
    float* adj  = xn + (size_t)BB * TT * DD;            // B*T*T
    int*   inds = (int*)(adj + (size_t)BB * TT * TT);   // B*T*2 ints
    float* rows = (float*)(inds + (size_t)BB * TT * 2); // B*T
    float* cols = rows + (size_t)BB * TT;               // B*T
    float* hA   = cols + (size_t)BB * TT;               // B*T*D
    float* hB   = hA + (size_t)BB * TT * DD;            // B*T*D
    float* h1   = hB + (size_t)BB * TT * DD;            // B*T*D

    const int gRows   = BB * TT;                                  // 32768
    const int gCorel  = (BB * (TT / 32) * (TT / 32)) / 8;         // 2048
    const int gTopk   = (BB * TT + 255) / 256;                    // 128
    const int gElemTT = (BB * TT * TT) / 256;                     // 65536
    const int gAdjX   = (BB * (TT / 32) * (DD / 32)) / 8;         // 1024
    const int gGemmW  = ((BB * TT / 32) * (DD / 32)) / 8;         // 1024

    // adjacency build
    k_rownorm <<<gRows,   256, 0, stream>>>(x, xn);
    k_corel   <<<gCorel,  256, 0, stream>>>(xn, sl, mask, adj);
    k_topk    <<<gTopk,   256, 0, stream>>>(adj, inds);
    k_adjbuild<<<gElemTT, 256, 0, stream>>>(mask, inds, adj);
    k_rowsum  <<<gRows,   256, 0, stream>>>(adj, rows);
    k_colsum  <<<gRows,   256, 0, stream>>>(adj, cols);
    k_adjnorm <<<gElemTT, 256, 0, stream>>>(adj, rows, cols, mask);

    // GCN layer 1
    k_adjx     <<<gAdjX,  256, 0, stream>>>(adj, x, hA);
    k_gemmwt   <<<gGemmW, 256, 0, stream>>>(hA, W1, b1, hB);
    k_layernorm<<<gRows,  256, 0, stream>>>(hB, g1, be1, h1);

    // GCN layer 2
    k_adjx     <<<gAdjX,  256, 0, stream>>>(adj, h1, hA);
    k_gemmwt   <<<gGemmW, 256, 0, stream>>>(hA, W2, b2, hB);
    k_layernorm<<<gRows,  256, 0, stream>>>(hB, g2, be2, out);
}